// ScaledDotProductAttention_73942156968083
// MI455X (gfx1250) — compile-verified
//
#include <hip/hip_runtime.h>

typedef float v2f __attribute__((ext_vector_type(2)));
typedef float v8f __attribute__((ext_vector_type(8)));

#define B_SZ 8192
#define T_SZ 64
#define D_SZ 512
#define INV_TEMP 0.04419417382415922f   /* 1/sqrt(512) */
#define MASK_FILL -1000.0f

// One workgroup (4 wave32) per batch row.
//   q staging:  GLOBAL_LOAD_ASYNC_TO_LDS_B128 (ASYNCcnt path, no VGPR round-trip)
//   Phase 1: scores = K(64x512) . q          via v_wmma_f32_16x16x4_f32
//   Softmax: masked, numerically stable, wave 0
//   Phase 2: out = attn(1x64) . V(64x512)    via v_wmma_f32_16x16x4_f32
__global__ __launch_bounds__(128) void sdpa_wmma_kernel(
    const float* __restrict__ q,        // [B, D]
    const float* __restrict__ k,        // [B, T, D]
    const float* __restrict__ v,        // [B, T, D]
    const int*   __restrict__ pad_mask, // [B, T]
    float* __restrict__ out,            // [B, D]
    float* __restrict__ attn_out)       // [B, T]
{
    __shared__ float q_lds[D_SZ];
    __shared__ float scores_lds[T_SZ];
    __shared__ float attn_lds[T_SZ];

    const int b    = blockIdx.x;
    const int tid  = threadIdx.x;
    const int wave = tid >> 5;       // 0..3
    const int lane = tid & 31;
    const int m    = lane & 15;      // row / column within 16
    const int g    = lane >> 4;      // 0: lanes 0-15 (K-slots 0,1), 1: lanes 16-31 (K-slots 2,3)

    // ---- stage q row into LDS: async global->LDS b128, tracked on ASYNCcnt ----
    // 128 threads * 16B = 512 floats = whole q row. Each wave drains its own
    // ASYNCcnt before signaling the barrier, so LDS is coherent after sync.
    {
        unsigned int      lds_off = (unsigned int)(size_t)q_lds + (unsigned int)tid * 16u;
        unsigned long long gaddr  = (unsigned long long)(size_t)(q + (size_t)b * D_SZ)
                                    + (unsigned long long)tid * 16ull;
        asm volatile("global_load_async_to_lds_b128 %0, %1, off"
                     :: "v"(lds_off), "v"(gaddr)
                     : "memory");
        asm volatile("s_wait_asynccnt 0" ::: "memory");
    }
    __syncthreads();

    // ---- Phase 1: wave 'wave' computes scores[t0 .. t0+15] ----
    // A-layout (16x4 f32): lanes 0-15 hold (K0,K1), lanes 16-31 hold (K2,K3), M = lane&15.
    // K-slots are relabeled so each lane loads one float4 of its K-row per 2 WMMAs:
    //   step (s,h): lanes<16 carry d = 8s+2h, 8s+2h+1 ; lanes>=16 carry d = 8s+4+2h, 8s+4+2h+1
    // B is the matching q chunk broadcast over all 16 columns -> every column of D = scores.
    const int    t0   = wave * 16;
    const float* krow = k + ((size_t)b * T_SZ + (t0 + m)) * D_SZ;

    v8f c = {0.f, 0.f, 0.f, 0.f, 0.f, 0.f, 0.f, 0.f};
#pragma unroll 4
    for (int s = 0; s < D_SZ / 8; ++s) {
        const int d4 = 8 * s + 4 * g;
        float4 kv = *(const float4*)(krow  + d4);   // global_load_b128, streams each K row
        float4 qv = *(const float4*)(q_lds + d4);   // ds_load_b128 broadcast
        v2f a0 = {kv.x, kv.y}, b0 = {qv.x, qv.y};
        c = __builtin_amdgcn_wmma_f32_16x16x4_f32(false, a0, false, b0, (short)0, c, false, false);
        v2f a1 = {kv.z, kv.w}, b1 = {qv.z, qv.w};
        c = __builtin_amdgcn_wmma_f32_16x16x4_f32(false, a1, false, b1, (short)0, c, false, false);
    }
    // D layout: VGPR r, lanes 0-15 -> M=r, lanes 16-31 -> M=8+r; all columns identical.
    if (lane == 0) {
#pragma unroll
        for (int r = 0; r < 8; ++r) scores_lds[t0 + r] = c[r];
    }
    if (lane == 16) {
#pragma unroll
        for (int r = 0; r < 8; ++r) scores_lds[t0 + 8 + r] = c[r];
    }
    __syncthreads();

    // ---- masked, stable softmax over T=64 (wave 0; 2 elements per lane) ----
    if (wave == 0) {
        const int ta = lane, tb = lane + 32;
        const int* mrow = pad_mask + (size_t)b * T_SZ;
        float s0 = scores_lds[ta] * INV_TEMP;
        float s1 = scores_lds[tb] * INV_TEMP;
        if (mrow[ta]) s0 = MASK_FILL;          // mask applied after scaling (matches reference)
        if (mrow[tb]) s1 = MASK_FILL;
        float mx = fmaxf(s0, s1);
#pragma unroll
        for (int off = 16; off > 0; off >>= 1)
            mx = fmaxf(mx, __shfl_xor(mx, off, 32));
        float e0 = __expf(s0 - mx);
        float e1 = __expf(s1 - mx);
        float sum = e0 + e1;
#pragma unroll
        for (int off = 16; off > 0; off >>= 1)
            sum += __shfl_xor(sum, off, 32);
        const float inv = 1.0f / sum;
        const float p0 = e0 * inv, p1 = e1 * inv;
        attn_lds[ta] = p0;
        attn_lds[tb] = p1;
        attn_out[(size_t)b * T_SZ + ta] = p0;
        attn_out[(size_t)b * T_SZ + tb] = p1;
    }
    __syncthreads();

    // ---- Phase 2: out[n] = sum_t attn[t] * v[t, n] ----
    // A rows are all identical (broadcast attn chunk); B = V tile in B-layout:
    // lane column n = lane&15, lanes<16 carry rows tc,tc+1; lanes>=16 carry rows tc+2,tc+3.
    // D columns n0..n0+15 appear on lanes 0-15 of c2[0]. V is read exactly once overall.
    const float* vbase = v + (size_t)b * T_SZ * D_SZ;
#pragma unroll 1
    for (int i = 0; i < 8; ++i) {
        const int n0 = 64 * i + 16 * wave;         // each wave covers 8 disjoint 16-wide tiles
        v8f c2 = {0.f, 0.f, 0.f, 0.f, 0.f, 0.f, 0.f, 0.f};
#pragma unroll
        for (int tc = 0; tc < T_SZ; tc += 4) {
            const int kr = tc + 2 * g;
            v2f a  = {attn_lds[kr], attn_lds[kr + 1]};
            v2f bb = {vbase[(size_t)kr * D_SZ + n0 + m],
                      vbase[(size_t)(kr + 1) * D_SZ + n0 + m]};
            c2 = __builtin_amdgcn_wmma_f32_16x16x4_f32(false, a, false, bb, (short)0, c2, false, false);
        }
        if (g == 0)
            out[(size_t)b * D_SZ + n0 + m] = c2[0];
    }
}

extern "C" void kernel_launch(void* const* d_in, const int* in_sizes, int n_in,
                              void* d_out, int out_size, void* d_ws, size_t ws_size,
                              hipStream_t stream) {
    const float* q        = (const float*)d_in[0];
    const float* k        = (const float*)d_in[1];
    const float* v        = (const float*)d_in[2];
    const int*   pad_mask = (const int*)d_in[3];

    float* out  = (float*)d_out;                          // [B,1,D] flat
    float* attn = (float*)d_out + (size_t)B_SZ * D_SZ;    // [B,1,T] flat, concatenated after out

    sdpa_wmma_kernel<<<B_SZ, 128, 0, stream>>>(q, k, v, pad_mask, out, attn);
}